// BSQ_970662609103
// MI455X (gfx1250) — compile-verified
//
#include <hip/hip_runtime.h>

// BSQ fused kernel for gfx1250 (MI455X).
//   out = sign(x @ Wp + bp) @ Wr + br
// (L2 normalization in the reference is sign-invariant -> dropped.)
//
// Both GEMMs run on the matrix pipe via V_WMMA_F32_16X16X4_F32 (full f32
// precision, no conversion). One wave32 handles one 16-pixel tile:
//   stage 1: z[16x16]   = x_tile[16x512] @ Wp[512x16]   (128 chained WMMAs)
//   stage 2: out[16x512]= sign(z)[16x16] @ Wr[16x512]   (32 N-blocks x 4 WMMAs)
// The 16x16 sign-code matrix is reshuffled across lanes through LDS.

typedef __attribute__((ext_vector_type(2))) float v2f;
typedef __attribute__((ext_vector_type(8))) float v8f;

#define WAVES_PER_BLOCK 4
#define C_DIM 512
#define BITS 16

__global__ __launch_bounds__(32 * WAVES_PER_BLOCK)
void bsq_fused_kernel(const float* __restrict__ x,
                      const float* __restrict__ Wp,
                      const float* __restrict__ bp,
                      const float* __restrict__ Wr,
                      const float* __restrict__ br,
                      float* __restrict__ out)
{
    __shared__ float codeLds[WAVES_PER_BLOCK][16][16];

    const int lane = threadIdx.x & 31;
    const int wave = threadIdx.x >> 5;
    const int half = lane >> 4;   // 0: lanes 0-15, 1: lanes 16-31
    const int m    = lane & 15;   // A-row (pixel) index / B-col index

    const int  tile = blockIdx.x * WAVES_PER_BLOCK + wave;
    const long p0   = (long)tile * 16;   // first pixel of this tile

    // ---------------- Stage 1: z = x_tile @ Wp ----------------
    // A frag (16x4 f32): lane holds M=m, K = 2*half + {0,1}  -> aligned 8B load
    // B frag (4x16 f32): lane holds K = 2*half + {0,1}, N = m
    const float* xrow   = x  + (p0 + m) * C_DIM + 2 * half;
    const float* wpbase = Wp + (2 * half) * BITS + m;

    v8f acc = {};
    #pragma unroll 4
    for (int k = 0; k < C_DIM / 4; ++k) {
        v2f a = *(const v2f*)(xrow + 4 * k);
        v2f b;
        b.x = wpbase[(4 * k + 0) * BITS];
        b.y = wpbase[(4 * k + 1) * BITS];
        acc = __builtin_amdgcn_wmma_f32_16x16x4_f32(
            /*neg_a=*/false, a, /*neg_b=*/false, b,
            /*c_mod=*/(short)0, acc, /*reuse_a=*/false, /*reuse_b=*/false);
    }

    // bias + sign -> stash 16x16 code matrix (+-1.0f) in LDS.
    // C layout: VGPR v, lane l  ->  (M = v + 8*half, N = m)
    const float bpv = bp[m];
    #pragma unroll
    for (int v = 0; v < 8; ++v) {
        const int   pix = v + 8 * half;
        const float z   = acc[v] + bpv;
        codeLds[wave][pix][m] = (z >= 0.0f) ? 1.0f : -1.0f;
    }
    __syncthreads();

    // ---------------- Stage 2: out = code @ Wr ----------------
    // A frags for K=16 (4 chained K=4 WMMAs), read once from LDS:
    // lane holds A[M=m][K = 4*kk + 2*half + {0,1}] -> aligned ds_load_b64
    v2f afr[4];
    #pragma unroll
    for (int kk = 0; kk < 4; ++kk)
        afr[kk] = *(const v2f*)&codeLds[wave][m][4 * kk + 2 * half];

    const float* wrbase  = Wr + (2 * half) * C_DIM + m;
    float* const outbase = out + p0 * C_DIM;

    for (int nb = 0; nb < C_DIM / 16; ++nb) {
        v8f acc2 = {};
        #pragma unroll
        for (int kk = 0; kk < 4; ++kk) {
            v2f b;
            b.x = wrbase[(4 * kk + 0) * C_DIM + nb * 16];
            b.y = wrbase[(4 * kk + 1) * C_DIM + nb * 16];
            acc2 = __builtin_amdgcn_wmma_f32_16x16x4_f32(
                false, afr[kk], false, b, (short)0, acc2, false, false);
        }
        const int   c   = nb * 16 + m;
        const float brv = br[c];
        #pragma unroll
        for (int v = 0; v < 8; ++v) {
            const int pix = v + 8 * half;
            outbase[(long)pix * C_DIM + c] = acc2[v] + brv;  // 64B-coalesced per half-wave
        }
    }
}

extern "C" void kernel_launch(void* const* d_in, const int* in_sizes, int n_in,
                              void* d_out, int out_size, void* d_ws, size_t ws_size,
                              hipStream_t stream) {
    (void)in_sizes; (void)n_in; (void)out_size; (void)d_ws; (void)ws_size;
    const float* x  = (const float*)d_in[0];  // [64,32,32,512]
    const float* Wp = (const float*)d_in[1];  // [512,16]
    const float* bp = (const float*)d_in[2];  // [16]
    const float* Wr = (const float*)d_in[3];  // [16,512]
    const float* br = (const float*)d_in[4];  // [512]
    float* out = (float*)d_out;               // [64,32,32,512]

    const int pixels = 64 * 32 * 32;                    // 65536
    const int tiles  = pixels / 16;                     // 4096
    const int blocks = tiles / WAVES_PER_BLOCK;         // 1024
    bsq_fused_kernel<<<blocks, 32 * WAVES_PER_BLOCK, 0, stream>>>(x, Wp, bp, Wr, br, out);
}